// SpikeFP32GELU_11450382811506
// MI455X (gfx1250) — compile-verified
//
#include <hip/hip_runtime.h>
#include <math.h>

// SpikeFP32GELU: decode 32-float bit pulses -> f32 -> FP64 GELU(x)=x*sigmoid(1.702x)
// -> f32 -> re-encode 32-float bit pulses. Memory-bound streaming kernel
// (536 MB traffic -> ~23us floor at 23.3 TB/s). Uses gfx1250 async global->LDS
// copies for coalesced input staging.

#define VALS_PER_BLOCK 256
#define LDS_VAL_STRIDE 144  // 128B payload + 16B pad: breaks the 16-way LDS bank conflict

typedef __attribute__((ext_vector_type(4))) unsigned int uint4v;
typedef __attribute__((ext_vector_type(4))) int int4v;
typedef __attribute__((ext_vector_type(4))) float float4v;

// Builtin signature (probe-confirmed by round-1 diagnostic): first arg is a
// pointer to v4i in the global address space, second a v4i* in LDS.
typedef __attribute__((address_space(1))) int4v g_int4v;
typedef __attribute__((address_space(3))) int4v l_int4v;

#if defined(__gfx1250__) && __has_builtin(__builtin_amdgcn_global_load_async_to_lds_b128)
#define HAVE_ASYNC_LDS 1
#else
#define HAVE_ASYNC_LDS 0
#endif

__global__ __launch_bounds__(VALS_PER_BLOCK) void spike_fp32_gelu_kernel(
    const float* __restrict__ x, float* __restrict__ out, int n_values) {
  __shared__ __align__(16) unsigned char lds_tile[VALS_PER_BLOCK * LDS_VAL_STRIDE];

  const int t = threadIdx.x;
  const int tile_base_val = (int)blockIdx.x * VALS_PER_BLOCK;

  // ---- Stage input tile (up to 256 values x 128B) into LDS, coalesced ----
  // chunk c = 16 bytes; consecutive threads take consecutive chunks so each
  // async-load instruction covers 512B of contiguous global memory per wave.
  const unsigned char* gbytes =
      (const unsigned char*)x + (size_t)tile_base_val * 128u;
  const int rem = n_values - tile_base_val;
  const int tile_vals = rem < VALS_PER_BLOCK ? rem : VALS_PER_BLOCK;
  const int tile_chunks = tile_vals * 8;

#pragma unroll
  for (int k = 0; k < 8; ++k) {
    const int c = k * VALS_PER_BLOCK + t;
    if (c < tile_chunks) {
      const int v = c >> 3;
      const int j = c & 7;
      const unsigned lds_off = (unsigned)v * LDS_VAL_STRIDE + (unsigned)j * 16u;
#if HAVE_ASYNC_LDS
      __builtin_amdgcn_global_load_async_to_lds_b128(
          (g_int4v*)(gbytes + (size_t)c * 16u),
          (l_int4v*)(lds_tile + lds_off),
          /*offset=*/0, /*cpol=*/0);
#else
      *(uint4v*)(lds_tile + lds_off) = *(const uint4v*)(gbytes + (size_t)c * 16u);
#endif
    }
  }
#if HAVE_ASYNC_LDS
#if __has_builtin(__builtin_amdgcn_s_wait_asynccnt)
  __builtin_amdgcn_s_wait_asynccnt(0);
#endif
#endif
  __syncthreads();

  const int my_val = tile_base_val + t;
  if (my_val >= n_values) return;

  // ---- Pack 32 pulse floats (MSB first) into a uint32 ----
  // Pulse floats are exactly 0.0f (0x00000000) or 1.0f (0x3F800000): bit30 is the bit.
  unsigned u = 0u;
  const unsigned char* slot = lds_tile + (unsigned)t * LDS_VAL_STRIDE;
#pragma unroll
  for (int i = 0; i < 8; ++i) {
    const uint4v q = *(const uint4v*)(slot + i * 16);
#pragma unroll
    for (int j = 0; j < 4; ++j) {
      u = (u << 1) | ((q[j] >> 30) & 1u);
    }
  }

  // ---- GELU in FP64 (matches reference rounding), back to FP32 ----
  const double v = (double)__uint_as_float(u);
  const double sig = 1.0 / (1.0 + exp(-1.702 * v));
  const unsigned r = __float_as_uint((float)(v * sig));

  // ---- Unpack result bits to 32 pulse floats; non-temporal stores (no reuse, 268MB > L2) ----
  float* optr = out + (size_t)my_val * 32u;
#pragma unroll
  for (int i = 0; i < 8; ++i) {
    float4v o;
#pragma unroll
    for (int j = 0; j < 4; ++j) {
      const unsigned bitpos = 31u - (unsigned)(i * 4 + j);
      o[j] = (float)((r >> bitpos) & 1u);
    }
    __builtin_nontemporal_store(o, (float4v*)(optr + i * 4));
  }
}

extern "C" void kernel_launch(void* const* d_in, const int* in_sizes, int n_in,
                              void* d_out, int out_size, void* d_ws, size_t ws_size,
                              hipStream_t stream) {
  const float* x = (const float*)d_in[0];
  float* out = (float*)d_out;
  const long long n_floats = (long long)in_sizes[0];  // 32*65536*32
  const int n_values = (int)(n_floats / 32);          // 2,097,152
  const int blocks = (n_values + VALS_PER_BLOCK - 1) / VALS_PER_BLOCK;
  spike_fp32_gelu_kernel<<<blocks, VALS_PER_BLOCK, 0, stream>>>(x, out, n_values);
}